// ConnectedWithKNN_72224170049742
// MI455X (gfx1250) — compile-verified
//
#include <hip/hip_runtime.h>
#include <float.h>

// Problem constants (from the reference)
#define G_   512
#define N_   512
#define K_   16
#define DIM_ 3

typedef __attribute__((ext_vector_type(2))) float v2f;
typedef __attribute__((ext_vector_type(8))) float v8f;

// Branchless sorted insertion into an ascending 16-entry (dist,idx) list.
// Fully unrolled, static indexing only -> stays in VGPRs.
__device__ __forceinline__ void knn_insert(float d, int i,
                                           float (&bd)[K_], int (&bi)[K_]) {
    float cd = d; int ci = i;
#pragma unroll
    for (int j = 0; j < K_; ++j) {
        bool c = cd < bd[j];
        float td = bd[j]; int ti = bi[j];
        bd[j] = c ? cd : td;  bi[j] = c ? ci : ti;
        cd    = c ? td : cd;  ci    = c ? ti : ci;
    }
}

__global__ __launch_bounds__(256)
void knn_wmma_kernel(const float* __restrict__ pos,
                     int* __restrict__ edge_out,
                     int* __restrict__ nedge_out) {
    // Positions + squared norms for this group, LDS-resident (8 KB).
    __shared__ float4 sp[N_];

    const int g   = blockIdx.x;
    const int tid = threadIdx.x;

    // Stage group's points: x,y,z and precomputed |p|^2 in .w
    const float* gp = pos + (size_t)g * N_ * DIM_;
    for (int p = tid; p < N_; p += 256) {
        float x = gp[p * 3 + 0];
        float y = gp[p * 3 + 1];
        float z = gp[p * 3 + 2];
        sp[p] = make_float4(x, y, z, x * x + y * y + z * z);
    }
    if (tid == 0) nedge_out[g] = N_ * K_;
    __syncthreads();

    const int lane = tid & 31;   // wave32 lane
    const int wave = tid >> 5;   // 8 waves per block
    const int lo   = lane & 15;
    const int hi   = lane >> 4;  // 0: rows 0-7 / K=0,1 ; 1: rows 8-15 / K=2,3

    // Each wave owns 4 query tiles of 16 rows: qt = wave, wave+8, wave+16, wave+24
    for (int j = 0; j < 4; ++j) {
        const int qt    = wave + 8 * j;
        const int qbase = qt * 16;

        // B operand (4x16): queries scaled by -2, with the padded K=3 row = 1.0
        // so the GEMM accumulates the candidate norm for free.
        // Lanes 0-15 carry K=0,1 (-2x,-2y); lanes 16-31 carry K=2,3 (-2z, 1).
        float4 q = sp[qbase + lo];
        v2f b;
        b.x = hi ? (-2.0f * q.z) : (-2.0f * q.x);
        b.y = hi ? 1.0f          : (-2.0f * q.y);
        const int query = qbase + lo;   // column index of this lane

        // Top-K lists hold PARTIAL distances: |c|^2 - 2 c.q  (the +|q|^2 term
        // is constant per query column -> order-preserving, so omitted).
        float bd[K_]; int bi[K_];
#pragma unroll
        for (int k = 0; k < K_; ++k) { bd[k] = FLT_MAX; bi[k] = 0; }

        // Sweep all candidate tiles.
        for (int ct = 0; ct < N_ / 16; ++ct) {
            const int cbase = ct * 16;

            // A operand (16x4): lanes 0-15 hold K=0,1 (x,y) for M=0..15;
            // lanes 16-31 hold K=2,3 (z, |c|^2).
            float4 cp = sp[cbase + lo];
            v2f a;
            a.x = hi ? cp.z : cp.x;
            a.y = hi ? cp.w : cp.y;

            v8f acc = {};
            // D[m][n] = |c_m|^2 - 2 * c_m . q_n   (distance up to +|q_n|^2)
            acc = __builtin_amdgcn_wmma_f32_16x16x4_f32(
                /*neg_a=*/false, a, /*neg_b=*/false, b,
                /*c_mod=*/(short)0, acc,
                /*reuse_a=*/false, /*reuse_b=*/false);

            // C/D layout: lane holds column n = lane%16, rows m = r + 8*(lane/16)
#pragma unroll
            for (int r = 0; r < 8; ++r) {
                const int   cidx = cbase + r + 8 * hi;
                const float dist = acc[r];
                if (cidx != query && dist < bd[K_ - 1])
                    knn_insert(dist, cidx, bd, bi);
            }
        }

        // Merge the two half-lists of each query column: lane l (<16) holds
        // candidates with (m mod 16) in [0,8), lane l+16 holds the rest.
#pragma unroll
        for (int k = 0; k < K_; ++k) {
            float od = __shfl_xor(bd[k], 16, 32);
            int   oi = __shfl_xor(bi[k], 16, 32);
            if (hi == 0 && od < bd[K_ - 1]) knn_insert(od, oi, bd, bi);
        }

        // Lanes 0-15 now hold the final sorted top-16 for their query.
        if (hi == 0) {
            const int from = g * N_ + query;
            int2* eo = (int2*)edge_out;
#pragma unroll
            for (int k = 0; k < K_; ++k) {
                eo[(size_t)from * K_ + k] = make_int2(from, g * N_ + bi[k]);
            }
        }
    }
}

extern "C" void kernel_launch(void* const* d_in, const int* in_sizes, int n_in,
                              void* d_out, int out_size, void* d_ws, size_t ws_size,
                              hipStream_t stream) {
    // d_in[0]: num_nodes (int32, G) -- constant N, unused on device
    // d_in[1]: positions (float32, G*N*DIM)
    const float* positions = (const float*)d_in[1];

    int* edge_out  = (int*)d_out;                         // (G*N*K, 2) int32
    int* nedge_out = edge_out + (size_t)G_ * N_ * K_ * 2; // (G,) int32

    knn_wmma_kernel<<<G_, 256, 0, stream>>>(positions, edge_out, nedge_out);
}